// My_Mamba_28991029248283
// MI455X (gfx1250) — compile-verified
//
#include <hip/hip_runtime.h>
#include <hip/hip_bf16.h>
#include <math.h>

// ---------------- problem constants ----------------
constexpr int Bc   = 4;
constexpr int Lc   = 60;
constexpr int Hc   = 768;
constexpr int Ic   = 1536;
constexpr int Nc   = 16;
constexpr int Rc   = 48;     // dt rank
constexpr int LYRc = 32;
constexpr int Kc   = 4;      // conv width
constexpr int DAc  = 128;
constexpr int RRc  = 4;      // attn heads
constexpr int INDIM = 13;
constexpr int ROWS  = Bc * Lc;   // 240 (multiple of 16)

typedef __attribute__((ext_vector_type(2))) float v2f;
typedef __attribute__((ext_vector_type(8))) float v8f;

// ---------------- device helpers ----------------
__device__ __forceinline__ float silu_f(float v) { return v / (1.f + __expf(-v)); }

__device__ __forceinline__ float act_apply(float v, int act) {
    switch (act) {
        case 1:  return v > 0.f ? v : 0.f;                       // relu
        case 2:  return (v > 20.f) ? v : log1pf(__expf(v));      // softplus
        case 3:  return tanhf(v);                                // tanh
        default: return v;
    }
}

// ---------------- WMMA f32 GEMM: C = act(A@W + bias [+ Cin]) ----------------
// One wave32 computes NT adjacent 16x16 tiles (shared A fragment, NT independent
// accumulator chains for ILP), K stepped by 4 via V_WMMA_F32_16X16X4_F32.
template <int NT>
__global__ void wmma_gemm_f32(const float* __restrict__ A, int lda,
                              const float* __restrict__ W, int ldw,
                              const float* __restrict__ bias,
                              const float* __restrict__ Cin, int ldci,
                              float* __restrict__ C, int ldc,
                              int M, int N, int K, int act) {
    const int lane   = threadIdx.x;                 // 0..31
    const int tileN0 = (blockIdx.x * blockDim.y + threadIdx.y) * NT;
    const int tileM  = blockIdx.y;
    const int tilesN = N >> 4;
    if (tileN0 + NT > tilesN) return;               // wave-uniform exit (EXEC all-1 for WMMA)

    const int sub   = lane & 15;
    const int hi    = lane >> 4;                    // 0 or 1
    const int khalf = hi * 2;                       // K sub-offset per ISA A/B layout
    const int arow  = tileM * 16 + sub;

    const float* ap = A + (size_t)arow * lda + khalf;
    const float* bp = W + (size_t)khalf * ldw + tileN0 * 16 + sub;

    v8f acc[NT];
#pragma unroll
    for (int t = 0; t < NT; ++t) acc[t] = {};

    for (int k = 0; k < K; k += 4) {
        v2f a = *(const v2f*)(ap + k);              // K=khalf, khalf+1 for this lane half
        v2f b[NT];
#pragma unroll
        for (int t = 0; t < NT; ++t) {
            b[t][0] = bp[(size_t)k * ldw + t * 16];
            b[t][1] = bp[(size_t)(k + 1) * ldw + t * 16];
        }
        if (k + 8 < K)                               // keep the weight stream ahead (near caches)
            __builtin_prefetch(bp + (size_t)(k + 8) * ldw, 0, 3);
#pragma unroll
        for (int t = 0; t < NT; ++t)
            acc[t] = __builtin_amdgcn_wmma_f32_16x16x4_f32(
                         false, a, false, b[t], (short)0, acc[t], false, false);
    }

    const int rbase = tileM * 16 + hi * 8;          // C/D layout: vgpr r -> M=r (+8 for hi half)
#pragma unroll
    for (int t = 0; t < NT; ++t) {
        const int col = (tileN0 + t) * 16 + sub;
        const float bv = bias ? bias[col] : 0.f;
#pragma unroll
        for (int r = 0; r < 8; ++r) {
            const int row = rbase + r;
            if (row >= M) continue;
            float v = acc[t][r] + bv;
            if (Cin) v += Cin[(size_t)row * ldci + col];
            C[(size_t)row * ldc + col] = act_apply(v, act);
        }
    }
}

// ---------------- scalar GEMM for awkward shapes (K=13, N=4, M=4) ----------------
__global__ void naive_gemm(const float* __restrict__ A, int lda,
                           const float* __restrict__ W, int ldw,
                           const float* __restrict__ bias,
                           float* __restrict__ C, int ldc,
                           int M, int N, int K, int act) {
    const int idx = blockIdx.x * blockDim.x + threadIdx.x;
    if (idx >= M * N) return;
    const int row = idx / N, col = idx % N;
    float acc = bias ? bias[col] : 0.f;
    const float* a = A + (size_t)row * lda;
    const float* w = W + col;
    for (int k = 0; k < K; ++k) acc += a[k] * w[(size_t)k * ldw];
    C[(size_t)row * ldc + col] = act_apply(acc, act);
}

// ---------------- RMSNorm over last dim (one block per row) ----------------
__global__ void rmsnorm_kernel(const float* __restrict__ x, const float* __restrict__ w,
                               float* __restrict__ out, int H) {
    const int row = blockIdx.x;
    const float* xr = x + (size_t)row * H;
    float s = 0.f;
    for (int i = threadIdx.x; i < H; i += blockDim.x) { float v = xr[i]; s += v * v; }
    __shared__ float red[256];
    red[threadIdx.x] = s;
    __syncthreads();
    for (int off = blockDim.x >> 1; off > 0; off >>= 1) {
        if (threadIdx.x < off) red[threadIdx.x] += red[threadIdx.x + off];
        __syncthreads();
    }
    const float scale = rsqrtf(red[0] / (float)H + 1e-5f);
    for (int i = threadIdx.x; i < H; i += blockDim.x)
        out[(size_t)row * H + i] = xr[i] * scale * w[i];
}

// ---------------- causal depthwise conv (K=4) + SiLU ----------------
__global__ void conv_silu_kernel(const float* __restrict__ hsg,   // (ROWS, 2I), hs = cols [0,I)
                                 const float* __restrict__ cw,    // (I, K)
                                 const float* __restrict__ cb,    // (I)
                                 float* __restrict__ hs) {        // (ROWS, I)
    const int idx = blockIdx.x * blockDim.x + threadIdx.x;
    if (idx >= ROWS * Ic) return;
    const int i  = idx % Ic;
    const int bl = idx / Ic;
    const int l  = bl % Lc;
    float acc = cb[i];
#pragma unroll
    for (int k = 0; k < Kc; ++k) {
        const int lt = l + k - (Kc - 1);
        if (lt >= 0)
            acc += hsg[(size_t)(bl + lt - l) * (2 * Ic) + i] * cw[i * Kc + k];
    }
    hs[idx] = silu_f(acc);
}

// ---------------- selective-scan: one thread per (b, i), N=16 state in regs ----------------
__global__ void scan_kernel(const float* __restrict__ dt,     // (ROWS, I) softplus-applied
                            const float* __restrict__ hs,     // (ROWS, I)
                            const float* __restrict__ sp,     // (ROWS, 80): B @ +48, C @ +64
                            const float* __restrict__ A_log,  // (I, N)
                            float* __restrict__ y) {          // (ROWS, I)
    int gid = blockIdx.x * blockDim.x + threadIdx.x;
    if (gid >= Bc * Ic) return;
    const int b = gid / Ic;
    const int i = gid % Ic;
    float a[Nc], st[Nc];
#pragma unroll
    for (int n = 0; n < Nc; ++n) {
        a[n]  = -__expf(A_log[(size_t)i * Nc + n]);
        st[n] = 0.f;
    }
    for (int t = 0; t < Lc; ++t) {
        const size_t r   = (size_t)(b * Lc + t);
        const float  dtv = dt[r * Ic + i];
        const float  xu  = dtv * hs[r * Ic + i];
        const float* spr = sp + r * (Rc + 2 * Nc);
        float acc = 0.f;
#pragma unroll
        for (int n = 0; n < Nc; ++n) {
            const float dA = __expf(dtv * a[n]);
            st[n] = dA * st[n] + xu * spr[Rc + n];
            acc  += st[n] * spr[Rc + Nc + n];
        }
        y[r * Ic + i] = acc;
    }
}

// ---------------- gating: y2 = (y + hs*Dp) * silu(gate) ----------------
__global__ void gate_kernel(const float* __restrict__ y, const float* __restrict__ hs,
                            const float* __restrict__ hsg, const float* __restrict__ Dp,
                            float* __restrict__ y2) {
    const int idx = blockIdx.x * blockDim.x + threadIdx.x;
    if (idx >= ROWS * Ic) return;
    const int i  = idx % Ic;
    const int bl = idx / Ic;
    const float g = hsg[(size_t)bl * (2 * Ic) + Ic + i];
    y2[idx] = (y[idx] + hs[idx] * Dp[i]) * silu_f(g);
}

// ---------------- softmax over L for each (b, r) ----------------
__global__ void softmax_L_kernel(float* __restrict__ s) {   // (ROWS, RR)
    const int b = blockIdx.x / RRc;
    const int r = blockIdx.x % RRc;
    const int t = threadIdx.x;                              // 64 threads
    __shared__ float red[64];
    float v = (t < Lc) ? s[(size_t)(b * Lc + t) * RRc + r] : -1e30f;
    red[t] = v; __syncthreads();
    for (int off = 32; off > 0; off >>= 1) {
        if (t < off) red[t] = fmaxf(red[t], red[t + off]);
        __syncthreads();
    }
    const float mx = red[0]; __syncthreads();
    float e = (t < Lc) ? __expf(v - mx) : 0.f;
    red[t] = e; __syncthreads();
    for (int off = 32; off > 0; off >>= 1) {
        if (t < off) red[t] += red[t + off];
        __syncthreads();
    }
    if (t < Lc) s[(size_t)(b * Lc + t) * RRc + r] = e / red[0];
}

// ---------------- attention pool: flat[b, r*H+h] = sum_l attn[b,l,r]*hn[b,l,h] ----------------
__global__ void pool_kernel(const float* __restrict__ attn, const float* __restrict__ hn,
                            float* __restrict__ flat) {
    const int idx = blockIdx.x * blockDim.x + threadIdx.x;
    if (idx >= Bc * RRc * Hc) return;
    const int h  = idx % Hc;
    const int br = idx / Hc;
    const int r  = br % RRc;
    const int b  = br / RRc;
    float acc = 0.f;
    for (int l = 0; l < Lc; ++l)
        acc += attn[(size_t)(b * Lc + l) * RRc + r] * hn[(size_t)(b * Lc + l) * Hc + h];
    flat[(size_t)b * (RRc * Hc) + r * Hc + h] = acc;
}

// ---------------- host launcher ----------------
static inline void launch_wmma(const float* A, int lda, const float* W, int ldw,
                               const float* bias, const float* Cin, int ldci,
                               float* C, int ldc, int M, int N, int K, int act,
                               hipStream_t s) {
    const int tilesN = N >> 4;
    dim3 blk(32, 4);
    if ((tilesN & 3) == 0) {
        const int waves = tilesN / 4;               // each wave covers 4 N-tiles
        dim3 grd((waves + 3) / 4, (M + 15) / 16);
        hipLaunchKernelGGL((wmma_gemm_f32<4>), grd, blk, 0, s,
                           A, lda, W, ldw, bias, Cin, ldci, C, ldc, M, N, K, act);
    } else {
        dim3 grd((tilesN + 3) / 4, (M + 15) / 16);
        hipLaunchKernelGGL((wmma_gemm_f32<1>), grd, blk, 0, s,
                           A, lda, W, ldw, bias, Cin, ldci, C, ldc, M, N, K, act);
    }
}

extern "C" void kernel_launch(void* const* d_in, const int* in_sizes, int n_in,
                              void* d_out, int out_size, void* d_ws, size_t ws_size,
                              hipStream_t stream) {
    const float* x      = (const float*)d_in[0];
    const float* iw1    = (const float*)d_in[1];
    const float* ib1    = (const float*)d_in[2];
    const float* iw2    = (const float*)d_in[3];
    const float* ib2    = (const float*)d_in[4];
    const float* iw3    = (const float*)d_in[5];
    const float* ib3    = (const float*)d_in[6];
    const float* norm_w = (const float*)d_in[7];
    const float* in_w   = (const float*)d_in[8];
    const float* conv_w = (const float*)d_in[9];
    const float* conv_b = (const float*)d_in[10];
    const float* xp_w   = (const float*)d_in[11];
    const float* dt_w   = (const float*)d_in[12];
    const float* dt_b   = (const float*)d_in[13];
    const float* A_log  = (const float*)d_in[14];
    const float* Dp     = (const float*)d_in[15];
    const float* out_w  = (const float*)d_in[16];
    const float* fnormw = (const float*)d_in[17];
    const float* ws1    = (const float*)d_in[18];
    const float* ws2    = (const float*)d_in[19];
    const float* ow1    = (const float*)d_in[20];
    const float* ob1    = (const float*)d_in[21];
    const float* ow2    = (const float*)d_in[22];
    const float* ob2    = (const float*)d_in[23];
    const float* ow3    = (const float*)d_in[24];
    const float* ob3    = (const float*)d_in[25];
    float* out = (float*)d_out;

    // ---- workspace bump allocator (all f32) ----
    float* ws = (float*)d_ws;
    size_t off = 0;
    auto alloc = [&](size_t n) { float* p = ws + off; off += n; return p; };
    float* cur  = alloc((size_t)ROWS * Hc);        // residual stream
    float* hn   = alloc((size_t)ROWS * Hc);        // normed activations
    float* hsg  = alloc((size_t)ROWS * 2 * Ic);    // in-proj out (hs | gate)
    float* hsb  = alloc((size_t)ROWS * Ic);        // conv+silu
    float* spb  = alloc((size_t)ROWS * (Rc + 2 * Nc));
    float* dtb_ = alloc((size_t)ROWS * Ic);        // softplus(dt)
    float* yb   = alloc((size_t)ROWS * Ic);        // scan output
    float* y2b  = alloc((size_t)ROWS * Ic);        // gated
    float* t1   = alloc((size_t)ROWS * 256);
    float* t2   = alloc((size_t)ROWS * 256);
    float* s1   = alloc((size_t)ROWS * DAc);
    float* s2   = alloc((size_t)ROWS * RRc);
    float* mfl  = alloc((size_t)Bc * RRc * Hc);
    float* o1   = alloc((size_t)Bc * 256);
    float* o2   = alloc((size_t)Bc * 256);
    (void)ws_size; (void)n_in; (void)in_sizes; (void)out_size;

    // ---- input MLP: 13 -> 256 -> 256 -> 768 ----
    naive_gemm<<<(ROWS * 256 + 255) / 256, 256, 0, stream>>>(
        x, INDIM, iw1, 256, ib1, t1, 256, ROWS, 256, INDIM, 1);
    launch_wmma(t1, 256, iw2, 256, ib2, nullptr, 0, t2, 256, ROWS, 256, 256, 1, stream);
    launch_wmma(t2, 256, iw3, Hc, ib3, nullptr, 0, cur, Hc, ROWS, Hc, 256, 0, stream);

    // ---- 32 Mamba layers ----
    for (int ly = 0; ly < LYRc; ++ly) {
        const float* nw  = norm_w + (size_t)ly * Hc;
        const float* inw = in_w   + (size_t)ly * Hc * 2 * Ic;
        const float* cw  = conv_w + (size_t)ly * Ic * Kc;
        const float* cb  = conv_b + (size_t)ly * Ic;
        const float* xpw = xp_w   + (size_t)ly * Ic * (Rc + 2 * Nc);
        const float* dtw = dt_w   + (size_t)ly * Rc * Ic;
        const float* dtbv= dt_b   + (size_t)ly * Ic;
        const float* al  = A_log  + (size_t)ly * Ic * Nc;
        const float* dp  = Dp     + (size_t)ly * Ic;
        const float* ow  = out_w  + (size_t)ly * Ic * Hc;

        rmsnorm_kernel<<<ROWS, 256, 0, stream>>>(cur, nw, hn, Hc);
        // in-proj (dominant GEMM): (240x768)@(768x3072)
        launch_wmma(hn, Hc, inw, 2 * Ic, nullptr, nullptr, 0, hsg, 2 * Ic,
                    ROWS, 2 * Ic, Hc, 0, stream);
        conv_silu_kernel<<<(ROWS * Ic + 255) / 256, 256, 0, stream>>>(hsg, cw, cb, hsb);
        // sp = hs @ xp_w : (240x1536)@(1536x80)   (5 N-tiles -> NT=1 path)
        launch_wmma(hsb, Ic, xpw, Rc + 2 * Nc, nullptr, nullptr, 0, spb, Rc + 2 * Nc,
                    ROWS, Rc + 2 * Nc, Ic, 0, stream);
        // dt = softplus(sp[:, :48] @ dt_w + dt_b), fused in WMMA epilogue
        launch_wmma(spb, Rc + 2 * Nc, dtw, Ic, dtbv, nullptr, 0, dtb_, Ic,
                    ROWS, Ic, Rc, 2, stream);
        scan_kernel<<<(Bc * Ic + 255) / 256, 256, 0, stream>>>(dtb_, hsb, spb, al, yb);
        gate_kernel<<<(ROWS * Ic + 255) / 256, 256, 0, stream>>>(yb, hsb, hsg, dp, y2b);
        // residual fused as WMMA C-input: cur = cur + y2 @ out_w
        launch_wmma(y2b, Ic, ow, Hc, nullptr, cur, Hc, cur, Hc, ROWS, Hc, Ic, 0, stream);
    }

    // ---- final norm + attention pooling + output MLP ----
    rmsnorm_kernel<<<ROWS, 256, 0, stream>>>(cur, fnormw, hn, Hc);
    // tanh(hn @ ws1) fused in WMMA epilogue: (240x768)@(768x128)
    launch_wmma(hn, Hc, ws1, DAc, nullptr, nullptr, 0, s1, DAc, ROWS, DAc, Hc, 3, stream);
    naive_gemm<<<(ROWS * RRc + 255) / 256, 256, 0, stream>>>(
        s1, DAc, ws2, RRc, nullptr, s2, RRc, ROWS, RRc, DAc, 0);
    softmax_L_kernel<<<Bc * RRc, 64, 0, stream>>>(s2);
    pool_kernel<<<(Bc * RRc * Hc + 255) / 256, 256, 0, stream>>>(s2, hn, mfl);
    naive_gemm<<<(Bc * 256 + 255) / 256, 256, 0, stream>>>(
        mfl, RRc * Hc, ow1, 256, ob1, o1, 256, Bc, 256, RRc * Hc, 1);
    naive_gemm<<<(Bc * 256 + 255) / 256, 256, 0, stream>>>(
        o1, 256, ow2, 256, ob2, o2, 256, Bc, 256, 256, 1);
    naive_gemm<<<1, 32, 0, stream>>>(
        o2, 256, ow3, 1, ob3, out, 1, Bc, 1, 256, 0);
}